// MultiHeadSelfAttention_41223096107785
// MI455X (gfx1250) — compile-verified
//
#include <hip/hip_runtime.h>
#include <math.h>
#include <stdint.h>

typedef __attribute__((ext_vector_type(2))) float v2f;
typedef __attribute__((ext_vector_type(8))) float v8f;

#define B_  2
#define S_  2048
#define E_  1024
#define H_  16
#define D_  64
#define HD_ 1024   // H_*D_

// Async global->LDS copy of 16 bytes per lane (CDNA5 ASYNCcnt path).
__device__ __forceinline__ void async_cp16(uint32_t lds_off, const float* g) {
  asm volatile("global_load_async_to_lds_b128 %0, %1, off"
               :: "v"(lds_off), "v"(g) : "memory");
}
__device__ __forceinline__ void async_wait0() {
  asm volatile("s_wait_asynccnt 0x0" ::: "memory");
}
__device__ __forceinline__ uint32_t lds_off32(const void* p) {
  return (uint32_t)(uintptr_t)p;   // low 32 bits of generic addr = LDS offset
}

// ---------------------------------------------------------------------------
// Generic fp32 WMMA GEMM: C[M,N] = A[M,K] @ W[K,N] + bias[N]
// block = 128 threads (4 waves); block tile 64x64; K staged in 16-wide chunks.
// A tile staged via async-to-LDS; W tile staged TRANSPOSED so every WMMA
// B-fragment is a contiguous ds_load_b64 (no repacking movs).
// ---------------------------------------------------------------------------
__global__ __launch_bounds__(128) void gemm_f32_wmma(
    const float* __restrict__ A, const float* __restrict__ W,
    const float* __restrict__ bias, float* __restrict__ C,
    int M, int N, int K)
{
  __shared__ float As [64 * 16];   // [m][k], 4 KB
  __shared__ float Bst[64 * 16];   // [n][k] (transposed), 4 KB

  const int tid  = threadIdx.x;
  const int wave = tid >> 5;
  const int lane = tid & 31;
  const int lm   = lane & 15;     // row of A-frag / col of B,C-frags
  const int lh   = lane >> 4;     // half-wave select (K-pair / M-half)
  const int m0   = blockIdx.x * 64;
  const int n0   = blockIdx.y * 64;

  v8f acc[4] = {};

  for (int k0 = 0; k0 < K; k0 += 16) {
    // A tile: 64x16 floats via async global->LDS b128 (2 per thread)
    #pragma unroll
    for (int i = 0; i < 2; ++i) {
      int idx = tid + i * 128;                  // f4 slot 0..255
      int ra = idx >> 2, ca = (idx & 3) << 2;
      async_cp16(lds_off32(&As[ra * 16 + ca]),
                 &A[(size_t)(m0 + ra) * K + k0 + ca]);
    }
    // W tile: 16x64 floats, stored transposed into Bst[n][k]
    #pragma unroll
    for (int i = 0; i < 2; ++i) {
      int idx = tid + i * 128;
      int kr = idx >> 4, nc = (idx & 15) << 2;
      float4 w4 = *(const float4*)&W[(size_t)(k0 + kr) * N + n0 + nc];
      Bst[(nc + 0) * 16 + kr] = w4.x;
      Bst[(nc + 1) * 16 + kr] = w4.y;
      Bst[(nc + 2) * 16 + kr] = w4.z;
      Bst[(nc + 3) * 16 + kr] = w4.w;
    }
    async_wait0();
    __syncthreads();
    #pragma unroll
    for (int kk = 0; kk < 4; ++kk) {
      v2f a = *(const v2f*)&As[(wave * 16 + lm) * 16 + kk * 4 + lh * 2];
      #pragma unroll
      for (int ct = 0; ct < 4; ++ct) {
        v2f b = *(const v2f*)&Bst[(ct * 16 + lm) * 16 + kk * 4 + lh * 2];
        acc[ct] = __builtin_amdgcn_wmma_f32_16x16x4_f32(
            false, a, false, b, (short)0, acc[ct], false, false);
      }
    }
    __syncthreads();
  }

  #pragma unroll
  for (int ct = 0; ct < 4; ++ct) {
    float bv = bias[n0 + ct * 16 + lm];
    #pragma unroll
    for (int i = 0; i < 8; ++i) {
      int r = i + 8 * lh;                      // C layout: VGPR i -> M=i / M=i+8
      C[(size_t)(m0 + wave * 16 + r) * N + n0 + ct * 16 + lm] = acc[ct][i] + bv;
    }
  }
}

// ---------------------------------------------------------------------------
// Fused attention: one block per (b, h, 16 query rows).
// Entire 16 x 2048 fp32 score panel lives in LDS (128 KB) -> exact masked
// softmax without HBM round-trips; attention matrix written exactly once.
// K/V chunks staged via async-to-LDS. LDS = 128 KB + 32 KB = 160 KB.
// ---------------------------------------------------------------------------
__global__ __launch_bounds__(256) void attn_fused(
    const float* __restrict__ q, const float* __restrict__ k,
    const float* __restrict__ v, const int* __restrict__ amask,
    float* __restrict__ a_out, float* __restrict__ o_int)
{
  __shared__ float sS[16 * S_];      // 128 KB scores/probs
  __shared__ float sKV[128 * 64];    // 32 KB K/V staging; aliased scratch
  float* sAux = sKV;                 // [0..15]=rowmax, [16..31]=denom (between phases)

  const int tid  = threadIdx.x;
  const int wave = tid >> 5;
  const int lane = tid & 31;
  const int lm   = lane & 15;
  const int lh   = lane >> 4;
  const int i0   = blockIdx.x * 16;
  const int h    = blockIdx.y;
  const int b    = blockIdx.z;

  const float* qb = q + ((size_t)(b * S_ + i0)) * HD_ + h * D_;
  const float* kb = k + (size_t)b * S_ * HD_ + h * D_;
  const float* vb = v + (size_t)b * S_ * HD_ + h * D_;
  const int*   mb = amask + b * S_;

  // Preload the 16 Q A-fragments (16x64 tile, constant for this block).
  v2f aq[16];
  #pragma unroll
  for (int kk = 0; kk < 16; ++kk)
    aq[kk] = *(const v2f*)&qb[(size_t)lm * HD_ + kk * 4 + lh * 2];

  // ---- pass 1: masked raw scores into sS ----
  for (int c0 = 0; c0 < S_; c0 += 128) {
    #pragma unroll
    for (int t = 0; t < 8; ++t) {
      int idx = tid + t * 256;
      int r = idx >> 4, c = (idx & 15) << 2;
      async_cp16(lds_off32(&sKV[r * 64 + c]),
                 &kb[(size_t)(c0 + r) * HD_ + c]);
    }
    async_wait0();
    __syncthreads();
    v8f acc = {};
    #pragma unroll
    for (int kk = 0; kk < 16; ++kk) {
      // B[kd][n] = K[c0 + wave*16 + n][kd]; pair contiguous in sKV row
      v2f bb = *(const v2f*)&sKV[(wave * 16 + lm) * 64 + kk * 4 + lh * 2];
      acc = __builtin_amdgcn_wmma_f32_16x16x4_f32(
          false, aq[kk], false, bb, (short)0, acc, false, false);
    }
    {
      int   c = c0 + wave * 16 + lm;
      float m = (float)mb[c];
      #pragma unroll
      for (int i = 0; i < 8; ++i) {
        int r = i + 8 * lh;
        sS[r * S_ + c] = acc[i] * 0.125f * m;   // 1/sqrt(64), then mask-mul
      }
    }
    __syncthreads();
  }

  // ---- pass 2: row max (16 threads per row, shuffle reduce) ----
  {
    int r = tid >> 4, seg = tid & 15;
    float mx = -__builtin_inff();
    for (int c = seg * 128; c < seg * 128 + 128; ++c)
      mx = fmaxf(mx, sS[r * S_ + c]);
    #pragma unroll
    for (int off = 8; off > 0; off >>= 1)
      mx = fmaxf(mx, __shfl_xor(mx, off, 16));
    if (seg == 0) sAux[r] = mx;
  }
  __syncthreads();

  // ---- pass 3: exp * mask, row sum, denom = sum + (sum==0) ----
  {
    int r = tid >> 4, seg = tid & 15;
    float mx = sAux[r];
    float ps = 0.f;
    for (int c = seg * 128; c < seg * 128 + 128; ++c) {
      float e = __expf(sS[r * S_ + c] - mx) * (float)mb[c];
      sS[r * S_ + c] = e;
      ps += e;
    }
    #pragma unroll
    for (int off = 8; off > 0; off >>= 1)
      ps += __shfl_xor(ps, off, 16);
    if (seg == 0) sAux[16 + r] = ps + (ps == 0.f ? 1.f : 0.f);
  }
  __syncthreads();

  // ---- pass 4: normalize; stream attention probs to HBM (written once) ----
  float* arow = a_out + (((size_t)(b * H_ + h)) * S_ + i0) * S_;
  for (int r = 0; r < 16; ++r) {
    float inv = 1.f / sAux[16 + r];
    for (int c = tid; c < S_; c += 256) {
      float p = sS[r * S_ + c] * inv;
      sS[r * S_ + c] = p;
      arow[(size_t)r * S_ + c] = p;
    }
  }
  __syncthreads();

  // ---- pass 5: O = P @ V ; k-dim split across waves, accumulate in regs ----
  v8f accv[4] = {};
  for (int c0 = 0; c0 < S_; c0 += 128) {
    #pragma unroll
    for (int t = 0; t < 8; ++t) {
      int idx = tid + t * 256;
      int r = idx >> 4, c = (idx & 15) << 2;
      async_cp16(lds_off32(&sKV[r * 64 + c]),
                 &vb[(size_t)(c0 + r) * HD_ + c]);
    }
    async_wait0();
    __syncthreads();
    #pragma unroll
    for (int kk = 0; kk < 4; ++kk) {
      int kc = c0 + wave * 16 + kk * 4 + lh * 2;   // global key index
      v2f a = *(const v2f*)&sS[lm * S_ + kc];
      int kr = wave * 16 + kk * 4 + lh * 2;        // row within V chunk
      #pragma unroll
      for (int ct = 0; ct < 4; ++ct) {
        v2f bb;
        bb.x = sKV[(kr + 0) * 64 + ct * 16 + lm];
        bb.y = sKV[(kr + 1) * 64 + ct * 16 + lm];
        accv[ct] = __builtin_amdgcn_wmma_f32_16x16x4_f32(
            false, a, false, bb, (short)0, accv[ct], false, false);
      }
    }
    __syncthreads();
  }

  // cross-wave reduction of partial O tiles via LDS float atomics (ds_add_f32)
  for (int i2 = tid; i2 < 16 * 64; i2 += 256) sKV[i2] = 0.f;
  __syncthreads();
  #pragma unroll
  for (int ct = 0; ct < 4; ++ct) {
    #pragma unroll
    for (int i = 0; i < 8; ++i) {
      int r = i + 8 * lh;
      atomicAdd(&sKV[r * 64 + ct * 16 + lm], accv[ct][i]);
    }
  }
  __syncthreads();
  float* ob = o_int + ((size_t)(b * S_ + i0)) * HD_ + h * D_;
  for (int i2 = tid; i2 < 16 * 64; i2 += 256) {
    int r = i2 >> 6, c = i2 & 63;
    ob[(size_t)r * HD_ + c] = sKV[i2];
  }
}

// ---------------------------------------------------------------------------
extern "C" void kernel_launch(void* const* d_in, const int* in_sizes, int n_in,
                              void* d_out, int out_size, void* d_ws, size_t ws_size,
                              hipStream_t stream)
{
  (void)in_sizes; (void)n_in; (void)out_size; (void)ws_size;

  const float* h  = (const float*)d_in[0];
  const int*   am = (const int*)  d_in[1];
  const float* Wq = (const float*)d_in[2];
  const float* bq = (const float*)d_in[3];
  const float* Wk = (const float*)d_in[4];
  const float* bk = (const float*)d_in[5];
  const float* Wv = (const float*)d_in[6];
  const float* bv = (const float*)d_in[7];
  const float* Wo = (const float*)d_in[8];
  const float* bo = (const float*)d_in[9];

  const int    M   = B_ * S_;           // 4096 rows
  const size_t QKV = (size_t)M * HD_;   // 4,194,304 elems (16 MB fp32)

  float* wq = (float*)d_ws;             // workspace: q,k,v,o_int (64 MB total)
  float* wk = wq + QKV;
  float* wv = wk + QKV;
  float* wo = wv + QKV;

  float* o_out = (float*)d_out;         // [B,S,E]
  float* a_out = o_out + QKV;           // [B,H,S,S]

  dim3 gg(M / 64, HD_ / 64);            // (64, 16)
  gemm_f32_wmma<<<gg, 128, 0, stream>>>(h, Wq, bq, wq, M, HD_, E_);
  gemm_f32_wmma<<<gg, 128, 0, stream>>>(h, Wk, bk, wk, M, HD_, E_);
  gemm_f32_wmma<<<gg, 128, 0, stream>>>(h, Wv, bv, wv, M, HD_, E_);

  dim3 ga(S_ / 16, H_, B_);             // (128, 16, 2)
  attn_fused<<<ga, 256, 0, stream>>>(wq, wk, wv, am, a_out, wo);

  dim3 go(M / 64, E_ / 64);             // (64, 16)
  gemm_f32_wmma<<<go, 128, 0, stream>>>(wo, Wo, bo, o_out, M, E_, HD_);
}